// GATLayer_32916629357433
// MI455X (gfx1250) — compile-verified
//
#include <hip/hip_runtime.h>
#include <math.h>

#define IN_DIM 128
#define OUT_DIM 64

typedef __attribute__((ext_vector_type(2))) float v2f;
typedef __attribute__((ext_vector_type(8))) float v8f;

// ---------------------------------------------------------------------------
// K1: z = h @ W^T via V_WMMA_F32_16X16X4_F32.
// One wave computes a 16(row)x16(col) tile of z; 4 waves/block cover all 64
// output columns of a 16-row strip. K=128 -> 32 chained WMMAs (fp32 accum).
// A-matrix 16x4 f32 layout: lanes 0-15 hold K=k0+0/k0+1 (v0/v1),
// lanes 16-31 hold K=k0+2/k0+3. B 4x16 mirrors it with lane&15 = column.
// C/D 16x16 f32: VGPR r = row r (lanes 0-15) and row r+8 (lanes 16-31).
// ---------------------------------------------------------------------------
__global__ __launch_bounds__(128) void gat_gemm_z(const float* __restrict__ h,
                                                  const float* __restrict__ W,
                                                  float* __restrict__ z,
                                                  int N) {
  const int wave  = threadIdx.x >> 5;       // 0..3 -> column tile
  const int lane  = threadIdx.x & 31;
  const int row0  = blockIdx.x << 4;
  const int col0  = wave << 4;
  const int l15   = lane & 15;
  const int khalf = (lane >> 4) << 1;       // 0 or 2

  int arow = row0 + l15;
  if (arow >= N) arow = N - 1;              // clamp: avoid OOB reads on tail
  const float* __restrict__ hrow = h + (size_t)arow * IN_DIM;
  const float* __restrict__ wrow = W + (size_t)(col0 + l15) * IN_DIM; // B col n = W row n

  v8f c = {};
#pragma unroll
  for (int k0 = 0; k0 < IN_DIM; k0 += 4) {
    v2f av, bv;
    av.x = hrow[k0 + khalf];
    av.y = hrow[k0 + khalf + 1];
    bv.x = wrow[k0 + khalf];
    bv.y = wrow[k0 + khalf + 1];
    // 8 args: (neg_a, A, neg_b, B, c_mod, C, reuse_a, reuse_b)
    c = __builtin_amdgcn_wmma_f32_16x16x4_f32(false, av, false, bv,
                                              (short)0, c, false, false);
  }

  const int n  = col0 + l15;
  const int mb = row0 + ((lane >> 4) << 3);
#pragma unroll
  for (int r = 0; r < 8; ++r) {
    const int m = mb + r;
    if (m < N) z[(size_t)m * OUT_DIM + n] = c[r];
  }
}

// ---------------------------------------------------------------------------
// K0: zero output accumulator, init segment-max to -inf, denom to 0.
// ---------------------------------------------------------------------------
__global__ void gat_init(float* __restrict__ out, float* __restrict__ m,
                         float* __restrict__ denom, int N) {
  const int i = blockIdx.x * blockDim.x + threadIdx.x;
  if (i < N * OUT_DIM) out[i] = 0.0f;
  if (i < N) {
    m[i]     = -INFINITY;
    denom[i] = 0.0f;
  }
}

// ---------------------------------------------------------------------------
// K2: s_src[n] = z[n] . a[:64], s_dst[n] = z[n] . a[64:]
// ---------------------------------------------------------------------------
__global__ void gat_scores(const float* __restrict__ z, const float* __restrict__ a,
                           float* __restrict__ s_src, float* __restrict__ s_dst, int N) {
  const int n = blockIdx.x * blockDim.x + threadIdx.x;
  if (n >= N) return;
  const float* __restrict__ zr = z + (size_t)n * OUT_DIM;
  float ss = 0.0f, sd = 0.0f;
#pragma unroll 8
  for (int j = 0; j < OUT_DIM; ++j) {
    const float v = zr[j];
    ss += v * a[j];
    sd += v * a[OUT_DIM + j];
  }
  s_src[n] = ss;
  s_dst[n] = sd;
}

// Order-preserving float max via integer atomics (no CAS loop; deterministic).
__device__ __forceinline__ void atomicMaxFloat(float* addr, float val) {
  const int bits = __float_as_int(val);
  if (bits >= 0)
    atomicMax((int*)addr, bits);                       // non-negative: signed max
  else
    atomicMin((unsigned int*)addr, (unsigned int)bits); // negative: unsigned min
}

// ---------------------------------------------------------------------------
// K3: e = leaky_relu(s_src[src] + s_dst[dst]); segment max into m[dst]
// ---------------------------------------------------------------------------
__global__ void gat_edge_max(const int* __restrict__ src, const int* __restrict__ dst,
                             const float* __restrict__ s_src, const float* __restrict__ s_dst,
                             float* __restrict__ ev, float* __restrict__ m, int E) {
  const int e = blockIdx.x * blockDim.x + threadIdx.x;
  if (e >= E) return;
  float v = s_src[src[e]] + s_dst[dst[e]];
  v = (v > 0.0f) ? v : 0.01f * v;           // leaky_relu, slope 0.01
  ev[e] = v;
  atomicMaxFloat(&m[dst[e]], v);
}

// ---------------------------------------------------------------------------
// K4: ee = exp(e - m[dst]); denom[dst] += ee  (ee overwrites ev in place)
// ---------------------------------------------------------------------------
__global__ void gat_edge_exp(const int* __restrict__ dst, const float* __restrict__ m,
                             float* __restrict__ ev, float* __restrict__ denom, int E) {
  const int e = blockIdx.x * blockDim.x + threadIdx.x;
  if (e >= E) return;
  const int d = dst[e];
  const float ee = expf(ev[e] - m[d]);
  ev[e] = ee;
  atomicAdd(&denom[d], ee);
}

// ---------------------------------------------------------------------------
// K5: out[dst] += alpha * z[src]   (64 consecutive threads handle one edge ->
// coalesced 256B gather of z[src] and coalesced atomics into the dst row)
// ---------------------------------------------------------------------------
__global__ __launch_bounds__(256) void gat_scatter(const int* __restrict__ src,
                                                   const int* __restrict__ dst,
                                                   const float* __restrict__ ee,
                                                   const float* __restrict__ denom,
                                                   const float* __restrict__ z,
                                                   float* __restrict__ out, int E) {
  const long tid = (long)blockIdx.x * 256 + threadIdx.x;
  const long e   = tid >> 6;       // edge index
  const int  j   = (int)(tid & 63);
  if (e >= E) return;
  const int d   = dst[e];
  const float den = denom[d];
  const float alpha = ee[e] / ((den == 0.0f) ? 1.0f : den);
  atomicAdd(&out[(size_t)d * OUT_DIM + j], alpha * z[(size_t)src[e] * OUT_DIM + j]);
}

// ---------------------------------------------------------------------------
// K6: ELU in place:  x > 0 ? x : exp(x)-1
// ---------------------------------------------------------------------------
__global__ void gat_elu(float* __restrict__ out, int total) {
  const int i = blockIdx.x * blockDim.x + threadIdx.x;
  if (i >= total) return;
  const float x = out[i];
  out[i] = (x > 0.0f) ? x : expm1f(x);
}

extern "C" void kernel_launch(void* const* d_in, const int* in_sizes, int n_in,
                              void* d_out, int out_size, void* d_ws, size_t ws_size,
                              hipStream_t stream) {
  const float* h   = (const float*)d_in[0];
  const int*   src = (const int*)d_in[1];
  const int*   dst = (const int*)d_in[2];
  const float* W   = (const float*)d_in[3];
  const float* a   = (const float*)d_in[4];
  float*       out = (float*)d_out;

  const int N = in_sizes[0] / IN_DIM;   // 100000
  const int E = in_sizes[1];            // 1600000

  // Workspace layout (floats): z | s_src | s_dst | m | denom | ev  (~33.6 MB)
  float* ws    = (float*)d_ws;
  float* z     = ws;
  float* s_src = z + (size_t)N * OUT_DIM;
  float* s_dst = s_src + N;
  float* m     = s_dst + N;
  float* denom = m + N;
  float* ev    = denom + N;

  const int tot = N * OUT_DIM;
  gat_init   <<<(tot + 255) / 256, 256, 0, stream>>>(out, m, denom, N);
  gat_gemm_z <<<(N + 15) / 16,     128, 0, stream>>>(h, W, z, N);
  gat_scores <<<(N + 255) / 256,   256, 0, stream>>>(z, a, s_src, s_dst, N);
  gat_edge_max<<<(E + 255) / 256,  256, 0, stream>>>(src, dst, s_src, s_dst, ev, m, E);
  gat_edge_exp<<<(E + 255) / 256,  256, 0, stream>>>(dst, m, ev, denom, E);
  const long scat = (long)E * OUT_DIM;
  gat_scatter<<<(unsigned)((scat + 255) / 256), 256, 0, stream>>>(src, dst, ev, denom, z, out, E);
  gat_elu    <<<(tot + 255) / 256, 256, 0, stream>>>(out, tot);
}